// DifferentiableRGBtoVel_6451040878728
// MI455X (gfx1250) — compile-verified
//
#include <hip/hip_runtime.h>

typedef float v2f __attribute__((ext_vector_type(2)));
typedef float v8f __attribute__((ext_vector_type(8)));

#define HW      262144   // 512*512
#define NPIX    1048576  // 4*512*512
#define NCOLORS 256
#define INV_T   100.0f   // 1 / TEMPERATURE

// One wave (32 lanes) handles 16 pixels against all 256 colormap entries.
// Logits s[p,k] = 2*img_p.c_k - |c_k|^2  (per-pixel |img|^2 cancels in softmax)
// computed as 16 x V_WMMA_F32_16X16X4_F32 (A = [r g b 1], B = [2r;2g;2b;-|c|^2]).
__global__ __launch_bounds__(256) void rgb2vel_wmma_kernel(
    const float* __restrict__ image,   // (4,3,512,512)
    const float* __restrict__ cmap,    // (256,3)
    const float* __restrict__ v_i,     // (256,)
    float* __restrict__ out)           // (4,512,512)
{
    __shared__ float ldsB[NCOLORS * 4];  // rows of B, packed per color: 2r,2g,2b,-|c|^2
    __shared__ float ldsV[NCOLORS];

    const int tid = threadIdx.x;
    if (tid < NCOLORS) {
        float r = cmap[tid * 3 + 0];
        float g = cmap[tid * 3 + 1];
        float b = cmap[tid * 3 + 2];
        ldsB[tid * 4 + 0] = 2.0f * r;
        ldsB[tid * 4 + 1] = 2.0f * g;
        ldsB[tid * 4 + 2] = 2.0f * b;
        ldsB[tid * 4 + 3] = -(r * r + g * g + b * b);
        ldsV[tid] = v_i[tid];
    }
    __syncthreads();

    const int lane    = tid & 31;
    const int half    = lane >> 4;        // 0: K={0,1}, 1: K={2,3} operand halves
    const int m       = lane & 15;        // pixel row / column-in-tile
    const int wave    = (blockIdx.x * blockDim.x + tid) >> 5;
    const int pixBase = wave << 4;        // 16 pixels per wave
    const int p       = pixBase + m;

    // image is (N=4, C=3, H*W); channel stride = HW
    const int n    = p >> 18;             // p / HW
    const int hw   = p & (HW - 1);
    const int rIdx = n * 3 * HW + hw;

    // A-matrix 16x4 per ISA layout: lanes 0-15 hold K=0,1 ; lanes 16-31 hold K=2,3
    v2f a;
    if (half == 0) {
        a.x = image[rIdx];            // K=0: r
        a.y = image[rIdx + HW];       // K=1: g
    } else {
        a.x = image[rIdx + 2 * HW];   // K=2: b
        a.y = 1.0f;                   // K=3: bias lane for -|c|^2
    }

    // 16 column tiles of 16 colors each -> all 256 logits for 16 pixels.
    v8f s[16];
#pragma unroll
    for (int t = 0; t < 16; ++t) {
        const int col = t * 16 + m;
        v2f bv;
        bv.x = ldsB[col * 4 + half * 2 + 0];  // K=0/2 row
        bv.y = ldsB[col * 4 + half * 2 + 1];  // K=1/3 row
        v8f zero = {};
        s[t] = __builtin_amdgcn_wmma_f32_16x16x4_f32(
            /*neg_a=*/false, a, /*neg_b=*/false, bv,
            /*c_mod=*/(short)0, zero, /*reuse_a=*/false, /*reuse_b=*/false);
    }

    // Row-wise max over all 256 columns: across tiles, then 16-lane butterfly.
    float mx[8];
#pragma unroll
    for (int j = 0; j < 8; ++j) {
        float v = s[0][j];
#pragma unroll
        for (int t = 1; t < 16; ++t) v = fmaxf(v, s[t][j]);
        mx[j] = v;
    }
#pragma unroll
    for (int off = 8; off >= 1; off >>= 1) {
#pragma unroll
        for (int j = 0; j < 8; ++j)
            mx[j] = fmaxf(mx[j], __shfl_xor(mx[j], off, 32));
    }

    // exp + weighted accumulation (softmax numerator/denominator).
    float sumE[8]  = {0, 0, 0, 0, 0, 0, 0, 0};
    float sumEV[8] = {0, 0, 0, 0, 0, 0, 0, 0};
#pragma unroll
    for (int t = 0; t < 16; ++t) {
        const float vcol = ldsV[t * 16 + m];  // v_i for this lane's column
#pragma unroll
        for (int j = 0; j < 8; ++j) {
            float e = __expf((s[t][j] - mx[j]) * INV_T);
            sumE[j]  += e;
            sumEV[j] += e * vcol;
        }
    }
#pragma unroll
    for (int off = 8; off >= 1; off >>= 1) {
#pragma unroll
        for (int j = 0; j < 8; ++j) {
            sumE[j]  += __shfl_xor(sumE[j],  off, 32);
            sumEV[j] += __shfl_xor(sumEV[j], off, 32);
        }
    }

    // Each 16-lane group now holds identical reduced values; lanes 0 and 16
    // write the 8 pixel results of their half.
    if (m == 0) {
#pragma unroll
        for (int j = 0; j < 8; ++j)
            out[pixBase + half * 8 + j] = sumEV[j] / sumE[j];
    }
}

extern "C" void kernel_launch(void* const* d_in, const int* in_sizes, int n_in,
                              void* d_out, int out_size, void* d_ws, size_t ws_size,
                              hipStream_t stream) {
    (void)in_sizes; (void)n_in; (void)d_ws; (void)ws_size; (void)out_size;
    const float* image = (const float*)d_in[0];  // 4*3*512*512 f32
    const float* cmap  = (const float*)d_in[1];  // 256*3 f32
    const float* v_i   = (const float*)d_in[2];  // 256 f32
    float* out = (float*)d_out;                  // 4*512*512 f32

    // 16 pixels per wave, 8 waves per block -> 128 pixels per block.
    const int blocks = NPIX / 128;               // 8192
    rgb2vel_wmma_kernel<<<blocks, 256, 0, stream>>>(image, cmap, v_i, out);
}